// SpectralDrone_69999376990526
// MI455X (gfx1250) — compile-verified
//
#include <hip/hip_runtime.h>
#include <stdint.h>
#include <math.h>

#define NS 176400           // samples per row
#define NB 256              // batch
#define CL 1050             // chunk length for IIR block-scan
#define NCH 168             // NS / CL
#define TILES 8             // samples-per-thread in synth kernel
#define SPB (TILES * 256)   // samples per block in synth kernel
#define TWOPI 6.2831853071795864769f

__constant__ float c_centers[5] = {100.0f, 350.0f, 900.0f, 2500.0f, 6000.0f};

// ---------------------------------------------------------------------------
// Biquad coefficients from raw params (denormalized inline)
// ---------------------------------------------------------------------------
__device__ __forceinline__ void biquad_coeffs(const float* __restrict__ params, int b,
                                              float& b0, float& b1, float& b2,
                                              float& a1, float& a2) {
    float fund = params[b * 14 + 9] * 360.0f + 40.0f;          // (40,400)
    float fq   = params[b * 14 + 11] * 9.5f + 0.5f;            // (0.5,10)
    float fc = fminf(fmaxf(fund, 20.0f), 22050.0f - 100.0f);
    float q  = fminf(fmaxf(fq, 0.5f), 15.0f);
    float w0 = TWOPI * fc / 44100.0f;
    float cw = __cosf(w0);
    float al = __sinf(w0) / (2.0f * q);
    float ia0 = 1.0f / (1.0f + al);
    b0 = (1.0f - cw) * 0.5f * ia0;
    b1 = (1.0f - cw) * ia0;
    b2 = b0;
    a1 = -2.0f * cw * ia0;
    a2 = (1.0f - al) * ia0;
}

// ---------------------------------------------------------------------------
// K1: additive synthesis + noise + envelope.  Noise streamed via CDNA5
// async global->LDS loads, double buffered; each lane consumes only its own
// LDS slot, so per-wave s_wait_asynccnt is the only sync needed.
// ---------------------------------------------------------------------------
__global__ void __launch_bounds__(256)
k_synth(const float* __restrict__ params, const float* __restrict__ noise,
        float* __restrict__ out) {
    __shared__ float smem[2][256];
    const int b   = blockIdx.y;
    const int tid = threadIdx.x;
    const int base_n = blockIdx.x * SPB;

    const float* prow = params + b * 14;
    float lvl[5];
#pragma unroll
    for (int i = 0; i < 5; ++i) lvl[i] = prow[i];              // (0,1) identity
    const float time_fade  = prow[5] * 2.0f - 1.0f;            // (-1,1)
    const float time_pulse = prow[6] * 5.0f;                   // (0,5)
    const float harm       = prow[7];                          // (0,1)
    const float detune     = prow[8] * 50.0f;                  // (0,50)
    const float fund       = prow[9] * 360.0f + 40.0f;         // (40,400)
    const float nlev       = prow[10] * 0.5f;                  // (0,0.5)
    const float attack     = prow[12] * 0.99f + 0.01f;         // (0.01,1)
    const float release    = prow[13] * 1.9f + 0.1f;           // (0.1,2)

    const float fs   = fund * 0.005f;                          // fundamental/200
    const float dodd = 1.0f + detune * (1.0f / 1200.0f);       // odd-harmonic detune
    float wb[5];
#pragma unroll
    for (int i = 0; i < 5; ++i) wb[i] = TWOPI * c_centers[i] * fs;

    const float an     = fmaxf(attack * 0.25f, 0.01f);
    const float rn     = release * 0.25f;
    const float rstart = 1.0f - rn;
    const float rden   = fmaxf(rn, 0.01f);
    const float dt     = 4.0f / 176399.0f;                     // linspace step

    const float* nbase = noise + (size_t)b * NS;

    // issue one async noise element per lane into LDS slot [buf][tid]
    auto issue = [&](int tile, int buf) {
        int n = base_n + tile * 256 + tid;
        n = min(n, NS - 1);                                    // clamp: keep EXEC full
        unsigned voff = (unsigned)n * 4u;
        unsigned lds  = (unsigned)(unsigned long long)(uintptr_t)&smem[buf][tid];
        asm volatile("global_load_async_to_lds_b32 %0, %1, %2"
                     :: "v"(lds), "v"(voff), "s"(nbase) : "memory");
    };

    issue(0, 0);
    for (int tile = 0; tile < TILES; ++tile) {
        const int buf = tile & 1;
        if (tile + 1 < TILES) {
            issue(tile + 1, buf ^ 1);
            asm volatile("s_wait_asynccnt 0x1" ::: "memory");
        } else {
            asm volatile("s_wait_asynccnt 0x0" ::: "memory");
        }
        const int n = base_n + tile * 256 + tid;
        const float nv = smem[buf][tid];
        const float t  = (float)n * dt;

        float acc = nv * nlev;
#pragma unroll
        for (int i = 0; i < 5; ++i) {
            float s = __sinf(wb[i] * t);
#pragma unroll
            for (int h = 2; h < 8; ++h) {
                float hf  = c_centers[i] * fs * (float)h;
                float d   = (h & 1) ? dodd : 1.0f;
                float wgt = (hf < 22050.0f) ? (harm / (float)h) : 0.0f;
                s += __sinf(wb[i] * ((float)h * d) * t) * wgt;
            }
            acc += s * lvl[i];
        }

        // time envelope
        const float tn = t * 0.25f;
        float ae  = fminf(fmaxf(tn / an, 0.0f), 1.0f);
        float re  = 1.0f - fminf(fmaxf((tn - rstart) / rden, 0.0f), 1.0f);
        float env = ae * re;
        env *= fminf(fmaxf(0.5f + 0.5f * tn * time_fade, 0.1f), 1.0f);
        env *= 0.5f + 0.5f * __cosf(TWOPI * time_pulse * t);
        env  = fminf(fmaxf(env, 0.01f), 1.0f);

        if (n < NS) out[(size_t)b * NS + n] = acc * env;
    }
}

// ---------------------------------------------------------------------------
// K2: per-(batch,chunk) particular solution (zero initial state) + chunk
// 2x2 transfer matrix from the two homogeneous responses. 3 independent
// FMA chains -> good ILP while parallel across 43008 threads.
// ---------------------------------------------------------------------------
__global__ void __launch_bounds__(256)
k_partial(const float* __restrict__ params, const float* __restrict__ audio,
          float* __restrict__ yp, float* __restrict__ mp) {
    const int idx = blockIdx.x * blockDim.x + threadIdx.x;
    if (idx >= NB * NCH) return;
    const int b = idx / NCH, c = idx % NCH;

    float b0, b1c, b2, a1, a2;
    biquad_coeffs(params, b, b0, b1c, b2, a1, a2);

    const size_t rowoff = (size_t)b * NS + (size_t)c * CL;
    float x1 = (c > 0) ? audio[rowoff - 1] : 0.0f;   // x history crosses chunks
    float x2 = (c > 0) ? audio[rowoff - 2] : 0.0f;
    float y1 = 0.0f, y2 = 0.0f;                      // particular: zero state
    float h1 = 1.0f, h2 = 0.0f;                      // homog resp to (1,0)
    float g1 = 0.0f, g2 = 1.0f;                      // homog resp to (0,1)

    for (int i = 0; i < CL; ++i) {
        float x  = audio[rowoff + i];
        float y  = b0 * x + b1c * x1 + b2 * x2 - a1 * y1 - a2 * y2;
        float hn = -a1 * h1 - a2 * h2;
        float gn = -a1 * g1 - a2 * g2;
        yp[rowoff + i] = y;
        x2 = x1; x1 = x; y2 = y1; y1 = y;
        h2 = h1; h1 = hn; g2 = g1; g1 = gn;
    }
    float* m = mp + (size_t)idx * 6;
    m[0] = h1; m[1] = g1;   // M row for y[L-1]
    m[2] = h2; m[3] = g2;   // M row for y[L-2]
    m[4] = y1; m[5] = y2;   // p
}

// ---------------------------------------------------------------------------
// K3: sequential composition of 2x2 chunk transfers (168 tiny steps per
// batch), emits each chunk's incoming state. Also zeroes per-batch max.
// ---------------------------------------------------------------------------
__global__ void k_scan(const float* __restrict__ mp, float* __restrict__ st,
                       unsigned* __restrict__ mx) {
    const int b = blockIdx.x * blockDim.x + threadIdx.x;
    if (b >= NB) return;
    mx[b] = 0u;
    float u = 0.0f, v = 0.0f;                        // (y[-1], y[-2])
    for (int c = 0; c < NCH; ++c) {
        const int idx = b * NCH + c;
        st[idx * 2 + 0] = u;
        st[idx * 2 + 1] = v;
        const float* m = mp + (size_t)idx * 6;
        float nu = m[0] * u + m[1] * v + m[4];
        float nv = m[2] * u + m[3] * v + m[5];
        u = nu; v = nv;
    }
}

// ---------------------------------------------------------------------------
// K4: fix-up y = y_p + u*h[n] + v*g[n]; fold in |y| max via uint atomicMax.
// ---------------------------------------------------------------------------
__global__ void __launch_bounds__(256)
k_fix(const float* __restrict__ params, const float* __restrict__ yp,
      const float* __restrict__ st, float* __restrict__ out,
      unsigned* __restrict__ mx) {
    const int idx = blockIdx.x * blockDim.x + threadIdx.x;
    if (idx >= NB * NCH) return;
    const int b = idx / NCH, c = idx % NCH;

    float b0, b1c, b2, a1, a2;
    biquad_coeffs(params, b, b0, b1c, b2, a1, a2);
    (void)b0; (void)b1c; (void)b2;

    const float u = st[idx * 2 + 0];
    const float v = st[idx * 2 + 1];
    float h1 = 1.0f, h2 = 0.0f, g1 = 0.0f, g2 = 1.0f;
    const size_t rowoff = (size_t)b * NS + (size_t)c * CL;
    float m = 0.0f;

    for (int i = 0; i < CL; ++i) {
        float hn = -a1 * h1 - a2 * h2;
        float gn = -a1 * g1 - a2 * g2;
        float y  = yp[rowoff + i] + u * hn + v * gn;
        out[rowoff + i] = y;
        m = fmaxf(m, fabsf(y));
        h2 = h1; h1 = hn; g2 = g1; g1 = gn;
    }
    atomicMax(mx + b, __float_as_uint(m));   // valid: m >= 0
}

// ---------------------------------------------------------------------------
// K5: normalize by per-batch max.
// ---------------------------------------------------------------------------
__global__ void __launch_bounds__(256)
k_norm(float* __restrict__ out, const unsigned* __restrict__ mx) {
    const long long i = (long long)blockIdx.x * blockDim.x + threadIdx.x;
    if (i >= (long long)NB * NS) return;
    const int b = (int)(i / NS);
    const float mv = fmaxf(__uint_as_float(mx[b]), 1e-8f);
    out[i] *= 0.9f / mv;
}

// ---------------------------------------------------------------------------
// Fallback: sequential per-row filter + normalize (used only if ws too small)
// ---------------------------------------------------------------------------
__global__ void k_seqfilter(const float* __restrict__ params, float* __restrict__ out) {
    const int b = blockIdx.x * blockDim.x + threadIdx.x;
    if (b >= NB) return;
    float b0, b1c, b2, a1, a2;
    biquad_coeffs(params, b, b0, b1c, b2, a1, a2);
    const size_t off = (size_t)b * NS;
    float x1 = 0, x2 = 0, y1 = 0, y2 = 0, m = 0;
    for (int n = 0; n < NS; ++n) {
        float x = out[off + n];
        float y = b0 * x + b1c * x1 + b2 * x2 - a1 * y1 - a2 * y2;
        out[off + n] = y;
        m = fmaxf(m, fabsf(y));
        x2 = x1; x1 = x; y2 = y1; y1 = y;
    }
    const float s = 0.9f / fmaxf(m, 1e-8f);
    for (int n = 0; n < NS; ++n) out[off + n] *= s;
}

// ---------------------------------------------------------------------------
extern "C" void kernel_launch(void* const* d_in, const int* in_sizes, int n_in,
                              void* d_out, int out_size, void* d_ws, size_t ws_size,
                              hipStream_t stream) {
    const float* params = (const float*)d_in[0];   // (256,14) f32
    const float* noise  = (const float*)d_in[1];   // (256,176400) f32
    float* out = (float*)d_out;                    // (256,176400) f32

    dim3 g1((NS + SPB - 1) / SPB, NB), blk(256);
    k_synth<<<g1, blk, 0, stream>>>(params, noise, out);

    const size_t YPF = (size_t)NB * NS;            // particular solution
    const size_t MPF = (size_t)NB * NCH * 6;       // transfer matrices
    const size_t STF = (size_t)NB * NCH * 2;       // per-chunk init states
    const size_t need = (YPF + MPF + STF) * sizeof(float) + NB * sizeof(unsigned);

    if (ws_size >= need) {
        float* yp = (float*)d_ws;
        float* mp = yp + YPF;
        float* st = mp + MPF;
        unsigned* mx = (unsigned*)(st + STF);
        const int nthreads = NB * NCH;             // 43008
        k_partial<<<(nthreads + 255) / 256, 256, 0, stream>>>(params, out, yp, mp);
        k_scan<<<1, 256, 0, stream>>>(mp, st, mx);
        k_fix<<<(nthreads + 255) / 256, 256, 0, stream>>>(params, yp, st, out, mx);
        k_norm<<<(int)(((long long)NB * NS + 255) / 256), 256, 0, stream>>>(out, mx);
    } else {
        k_seqfilter<<<8, 32, 0, stream>>>(params, out);
    }
}